// LocalPNet_54425825575435
// MI455X (gfx1250) — compile-verified
//
#include <hip/hip_runtime.h>
#include <hip/hip_bf16.h>

typedef __attribute__((ext_vector_type(2))) float v2f;
typedef __attribute__((ext_vector_type(8))) float v8f;

#define HID 32

// ---------------------------------------------------------------------------
// Degree / normalization
// ---------------------------------------------------------------------------
__global__ void deg_init_kernel(float* __restrict__ deg, int n) {
    int i = blockIdx.x * blockDim.x + threadIdx.x;
    if (i < n) deg[i] = 1.0f;  // +1 self-loop
}

__global__ void deg_accum_kernel(const int* __restrict__ dst, float* __restrict__ deg, int e) {
    int i = blockIdx.x * blockDim.x + threadIdx.x;
    if (i < e) unsafeAtomicAdd(&deg[dst[i]], 1.0f);
}

__global__ void rsqrt_kernel(float* __restrict__ deg, int n) {
    int i = blockIdx.x * blockDim.x + threadIdx.x;
    if (i < n) deg[i] = rsqrtf(deg[i]);  // deg -> dinv in place
}

// ---------------------------------------------------------------------------
// WMMA GEMM: out[N,32] = in[N,KDIM] @ W[KDIM,32]   (f32, exact)
// One wave = 16 rows x 32 cols = two 16x16 f32 accumulators.
// V_WMMA_F32_16X16X4_F32, K accumulated in steps of 4.
// ---------------------------------------------------------------------------
template <int KDIM>
__global__ __launch_bounds__(256) void gemm_wmma_kernel(const float* __restrict__ in,
                                                        const float* __restrict__ W,
                                                        float* __restrict__ out, int n) {
    // W staged transposed in LDS: ldsW[col][k] so B-fragments are 8B-contiguous
    __shared__ float ldsW[HID * KDIM];
    const int tid = threadIdx.x;
    for (int idx = tid; idx < KDIM * HID; idx += 256) {
        int k = idx >> 5, c = idx & 31;
        ldsW[c * KDIM + k] = W[idx];
    }
    __syncthreads();

    const int lane = tid & 31;
    const int wave = tid >> 5;
    const int half = lane >> 4;   // 0: K pair {k,k+1}, 1: {k+2,k+3}
    const int l16  = lane & 15;   // M index (A) / N index (B,C)

    const int rbase = (blockIdx.x * 8 + wave) * 16;
    int arow = rbase + l16;
    if (arow >= n) arow = n - 1;  // clamp: EXEC must be all-1s for WMMA
    const float* arowp = in + (size_t)arow * KDIM;

    v8f acc0 = {};
    v8f acc1 = {};
#pragma unroll
    for (int k = 0; k < KDIM; k += 4) {
        const int kk = k + half * 2;
        v2f a  = *(const v2f*)(arowp + kk);
        v2f b0 = *(const v2f*)(&ldsW[(l16 + 0)  * KDIM + kk]);
        v2f b1 = *(const v2f*)(&ldsW[(l16 + 16) * KDIM + kk]);
        acc0 = __builtin_amdgcn_wmma_f32_16x16x4_f32(false, a, false, b0,
                                                     (short)0, acc0, false, false);
        acc1 = __builtin_amdgcn_wmma_f32_16x16x4_f32(false, a, false, b1,
                                                     (short)0, acc1, false, false);
    }

    // C/D layout: VGPR r -> M = r (lanes 0-15) / r+8 (lanes 16-31), N = l16
#pragma unroll
    for (int r = 0; r < 8; ++r) {
        int row = rbase + r + half * 8;
        if (row < n) {
            out[(size_t)row * HID + l16]      = acc0[r];
            out[(size_t)row * HID + l16 + 16] = acc1[r];
        }
    }
}

// ---------------------------------------------------------------------------
// GCN aggregation (32-wide features)
// ---------------------------------------------------------------------------
// out[i,:] = h[i,:]*dinv[i]^2 + b   (self-loop term + bias, accumulator init)
__global__ void node_init_feat_kernel(const float* __restrict__ h,
                                      const float* __restrict__ dinv,
                                      const float* __restrict__ b,
                                      float* __restrict__ out, int n) {
    int t = blockIdx.x * blockDim.x + threadIdx.x;
    int i = t >> 5, j = t & 31;
    if (i >= n) return;
    float di = dinv[i];
    out[(size_t)i * HID + j] = h[(size_t)i * HID + j] * di * di + b[j];
}

// out[dst,:] += h[src,:] * dinv[src]*dinv[dst]   (one wave per edge, coalesced)
__global__ void edge_agg_feat_kernel(const int* __restrict__ src,
                                     const int* __restrict__ dst,
                                     const float* __restrict__ dinv,
                                     const float* __restrict__ h,
                                     float* __restrict__ out, int e) {
    int t = blockIdx.x * blockDim.x + threadIdx.x;
    int ei = t >> 5, j = t & 31;
    if (ei >= e) return;
    int s = src[ei], d = dst[ei];
    float en = dinv[s] * dinv[d];
    float v = h[(size_t)s * HID + j] * en;
    unsafeAtomicAdd(&out[(size_t)d * HID + j], v);
}

// ---------------------------------------------------------------------------
// Scalar heads: s_mu = relu(h2)@W2, s_std = h2@Ws; init outputs with self term
// ---------------------------------------------------------------------------
__global__ void scalar_head_kernel(const float* __restrict__ h2,
                                   const float* __restrict__ dinv,
                                   const float* __restrict__ W2, const float* __restrict__ b2,
                                   const float* __restrict__ Ws, const float* __restrict__ bs,
                                   float* __restrict__ s_mu, float* __restrict__ s_std,
                                   float* __restrict__ mu, float* __restrict__ stdv, int n) {
    int i = blockIdx.x * blockDim.x + threadIdx.x;
    if (i >= n) return;
    const float* hp = h2 + (size_t)i * HID;
    float smu = 0.0f, sst = 0.0f;
#pragma unroll
    for (int j = 0; j < HID; ++j) {
        float hv = hp[j];
        smu = fmaf(fmaxf(hv, 0.0f), W2[j], smu);
        sst = fmaf(hv, Ws[j], sst);
    }
    s_mu[i]  = smu;
    s_std[i] = sst;
    float di = dinv[i];
    float sn = di * di;
    mu[i]   = b2[0] + smu * sn;
    stdv[i] = bs[0] + sst * sn;
}

__global__ void edge_agg_scalar_kernel(const int* __restrict__ src,
                                       const int* __restrict__ dst,
                                       const float* __restrict__ dinv,
                                       const float* __restrict__ s_mu,
                                       const float* __restrict__ s_std,
                                       float* __restrict__ mu, float* __restrict__ stdv, int e) {
    int i = blockIdx.x * blockDim.x + threadIdx.x;
    if (i >= e) return;
    int s = src[i], d = dst[i];
    float en = dinv[s] * dinv[d];
    unsafeAtomicAdd(&mu[d],   s_mu[s]  * en);
    unsafeAtomicAdd(&stdv[d], s_std[s] * en);
}

// ---------------------------------------------------------------------------
// Launch
// ---------------------------------------------------------------------------
extern "C" void kernel_launch(void* const* d_in, const int* in_sizes, int n_in,
                              void* d_out, int out_size, void* d_ws, size_t ws_size,
                              hipStream_t stream) {
    const float* x   = (const float*)d_in[0];
    const int*   ei  = (const int*)d_in[1];
    const float* W1  = (const float*)d_in[2];
    const float* b1  = (const float*)d_in[3];
    const float* Wh  = (const float*)d_in[4];
    const float* bh  = (const float*)d_in[5];
    const float* W2  = (const float*)d_in[6];
    const float* b2  = (const float*)d_in[7];
    const float* Ws  = (const float*)d_in[8];
    const float* bs  = (const float*)d_in[9];

    const int N = in_sizes[0] / 128;   // 100000
    const int E = in_sizes[1] / 2;     // 1600000
    const int* src = ei;               // edge_index[0]
    const int* dst = ei + E;           // edge_index[1]

    float* mu   = (float*)d_out;       // [N]
    float* stdv = mu + N;              // [N]

    // workspace layout
    float* dinv  = (float*)d_ws;                 // N  (deg, then dinv in place)
    float* bufA  = dinv + N;                     // N*32
    float* bufB  = bufA + (size_t)N * HID;       // N*32
    float* s_mu  = bufB + (size_t)N * HID;       // N
    float* s_std = s_mu + N;                     // N

    const int T = 256;
    dim3 blk(T);
    const int gN   = (N + T - 1) / T;            // node-parallel
    const int gE   = (E + T - 1) / T;            // edge-parallel
    const int gN32 = ((size_t)N * HID + T - 1) / T;  // node x feature
    const int gE32 = ((size_t)E * HID + T - 1) / T;  // edge x feature (wave/edge)
    const int gG   = (N + 127) / 128;            // 8 waves x 16 rows per block

    // normalization
    deg_init_kernel<<<gN, blk, 0, stream>>>(dinv, N);
    deg_accum_kernel<<<gE, blk, 0, stream>>>(dst, dinv, E);
    rsqrt_kernel<<<gN, blk, 0, stream>>>(dinv, N);

    // layer 1: h1 = GCN(x @ W1 + b1)
    gemm_wmma_kernel<128><<<gG, blk, 0, stream>>>(x, W1, bufA, N);
    node_init_feat_kernel<<<gN32, blk, 0, stream>>>(bufA, dinv, b1, bufB, N);
    edge_agg_feat_kernel<<<gE32, blk, 0, stream>>>(src, dst, dinv, bufA, bufB, E);

    // layer 2: h2 = GCN(h1 @ Wh + bh)
    gemm_wmma_kernel<32><<<gG, blk, 0, stream>>>(bufB, Wh, bufA, N);
    node_init_feat_kernel<<<gN32, blk, 0, stream>>>(bufA, dinv, bh, bufB, N);
    edge_agg_feat_kernel<<<gE32, blk, 0, stream>>>(src, dst, dinv, bufA, bufB, E);

    // heads: mu = GCN(relu(h2) @ W2 + b2), std = GCN(h2 @ Ws + bs)
    scalar_head_kernel<<<gN, blk, 0, stream>>>(bufB, dinv, W2, b2, Ws, bs,
                                               s_mu, s_std, mu, stdv, N);
    edge_agg_scalar_kernel<<<gE, blk, 0, stream>>>(src, dst, dinv, s_mu, s_std,
                                                   mu, stdv, E);
}